// RunLoRALinear_85401129714088
// MI455X (gfx1250) — compile-verified
//
#include <hip/hip_runtime.h>

// Fused LoRA linear for MI455X (gfx1250, wave32):
//   out = x@W + (x@(U*2))@V + bias
// Memory-bound (≈320 MB HBM traffic → ~14 us at 23.3 TB/s), so we keep full
// fp32 precision and use V_WMMA_F32_16X16X4_F32 for all matrix math.

typedef float v2f __attribute__((ext_vector_type(2)));
typedef float v4f __attribute__((ext_vector_type(4)));
typedef float v8f __attribute__((ext_vector_type(8)));

#define LORA_SCALING 2.0f

constexpr int DIN    = 4096;
constexpr int DOUT   = 4096;
constexpr int MROWS  = 4 * 2048;   // B*S
constexpr int RANK   = 16;

constexpr int M_TILE = 128;
constexpr int N_TILE = 128;
constexpr int K_TILE = 32;
constexpr int LDA    = 36;         // padded k-stride (floats): 36*4=144B, 16B-aligned rows

// D = A(16x4) * B(4x16) + C, fp32 WMMA.
// A frag (per lane): lanes 0-15 -> M=lane, {K=k,K=k+1}; lanes 16-31 -> {K=k+2,K=k+3}
// B frag (per lane): lanes 0-15 -> N=lane, {K=k,K=k+1}; lanes 16-31 -> {K=k+2,K=k+3}
// C/D: lane%16 = N, vgpr i -> M = i + (lane>=16 ? 8 : 0)
__device__ __forceinline__ v8f wmma_f32(v2f a, v2f b, v8f c) {
  return __builtin_amdgcn_wmma_f32_16x16x4_f32(
      /*neg_a=*/false, a, /*neg_b=*/false, b,
      /*c_mod=*/(short)0, c, /*reuse_a=*/false, /*reuse_b=*/false);
}

__global__ __launch_bounds__(256)
void lora_linear_fused(const float* __restrict__ x,
                       const float* __restrict__ W,
                       const float* __restrict__ U,
                       const float* __restrict__ V,
                       const float* __restrict__ bias,
                       float* __restrict__ out)
{
  __shared__ float A_lds[M_TILE * LDA];     // x tile, [m][k]
  __shared__ float B_lds[N_TILE * LDA];     // W tile, transposed to [n][k]
  __shared__ float U_lds[RANK * LDA];       // U tile, transposed to [r][k]
  __shared__ float low_lds[M_TILE * RANK];  // low = (x@U)*scaling for this M tile

  const int t    = threadIdx.x;
  const int lane = t & 31;
  const int wave = t >> 5;
  const int wm   = wave & 3;     // wave row    (4 along M -> 32 rows each)
  const int wn   = wave >> 2;    // wave column (2 along N -> 64 cols each)
  const int kh   = lane >> 4;    // half-wave select (K+0/1 vs K+2/3)
  const int l16  = lane & 15;

  const int n0 = blockIdx.x * N_TILE;
  const int m0 = blockIdx.y * M_TILE;

  const v8f vzero = {0.f, 0.f, 0.f, 0.f, 0.f, 0.f, 0.f, 0.f};
  v8f acc[2][4];                 // 2 M-subtiles x 4 N-subtiles of 16x16
  v8f lacc[2];                   // low accumulators (M x RANK)
#pragma unroll
  for (int s = 0; s < 2; ++s) {
    lacc[s] = vzero;
#pragma unroll
    for (int j = 0; j < 4; ++j) acc[s][j] = vzero;
  }

  for (int kb = 0; kb < DIN; kb += K_TILE) {
    // ---- stage x tile: 128x32 floats, 4 x float4 per thread ----
#pragma unroll
    for (int jj = 0; jj < 4; ++jj) {
      const int idx = t + jj * 256;
      const int row = idx >> 3;          // 0..127
      const int c4  = idx & 7;           // 0..7 (float4 column)
      v4f a4 = *(const v4f*)&x[(size_t)(m0 + row) * DIN + kb + c4 * 4];
      *(v4f*)&A_lds[row * LDA + c4 * 4] = a4;
    }
    // ---- stage W tile transposed: 32(K) x 128(N) -> B_lds[n][k] ----
#pragma unroll
    for (int jj = 0; jj < 4; ++jj) {
      const int idx = t + jj * 256;
      const int kl  = idx >> 5;          // 0..31
      const int c4  = idx & 31;          // 0..31 (float4 column in N)
      v4f w4 = *(const v4f*)&W[(size_t)(kb + kl) * DOUT + n0 + c4 * 4];
#pragma unroll
      for (int e = 0; e < 4; ++e)
        B_lds[(c4 * 4 + e) * LDA + kl] = w4[e];
    }
    // ---- stage U tile transposed: 32(K) x 16(R) -> U_lds[r][k] ----
#pragma unroll
    for (int jj = 0; jj < 2; ++jj) {
      const int idx = t + jj * 256;      // 0..511
      const int kl  = idx & 31;
      const int r   = idx >> 5;          // 0..15
      U_lds[r * LDA + kl] = U[(size_t)(kb + kl) * RANK + r];
    }
    // prefetch next K-stage of x and W into caches (global_prefetch_b8)
    const int kb2 = kb + K_TILE;
    if (kb2 < DIN) {
      __builtin_prefetch(&x[(size_t)(m0 + (t & 127)) * DIN + kb2 + (t >> 7) * 16], 0, 1);
      __builtin_prefetch(&W[(size_t)(kb2 + (t & 31)) * DOUT + n0 + (t >> 5) * 16], 0, 1);
    }
    __syncthreads();

    // ---- compute: 8 k-steps of depth 4; 8 GEMM WMMAs + 2 LoRA WMMAs each ----
#pragma unroll
    for (int k = 0; k < K_TILE; k += 4) {
      const int kk = k + kh * 2;
      v2f a[2], b[4], u;
#pragma unroll
      for (int s = 0; s < 2; ++s)
        a[s] = *(const v2f*)&A_lds[(wm * 32 + s * 16 + l16) * LDA + kk];
#pragma unroll
      for (int j = 0; j < 4; ++j)
        b[j] = *(const v2f*)&B_lds[(wn * 64 + j * 16 + l16) * LDA + kk];
      u = *(const v2f*)&U_lds[l16 * LDA + kk];
#pragma unroll
      for (int s = 0; s < 2; ++s) {
#pragma unroll
        for (int j = 0; j < 4; ++j)
          acc[s][j] = wmma_f32(a[s], b[j], acc[s][j]);
        lacc[s] = wmma_f32(a[s], u, lacc[s]);
      }
    }
    __syncthreads();
  }

  // ---- C-layout -> A-layout transpose of low via LDS (scaled) ----
  if (wn == 0) {
#pragma unroll
    for (int s = 0; s < 2; ++s) {
#pragma unroll
      for (int i = 0; i < 8; ++i) {
        const int ml = wm * 32 + s * 16 + i + kh * 8;   // local M row
        low_lds[ml * RANK + l16] = lacc[s][i] * LORA_SCALING;
      }
    }
  }
  __syncthreads();

  // ---- fold low@V into the same accumulators: 4 WMMA k-steps (K = RANK = 16) ----
#pragma unroll
  for (int k = 0; k < RANK; k += 4) {
    const int kk = k + kh * 2;
    v2f a[2], b[4];
#pragma unroll
    for (int s = 0; s < 2; ++s)
      a[s] = *(const v2f*)&low_lds[(wm * 32 + s * 16 + l16) * RANK + kk];
#pragma unroll
    for (int j = 0; j < 4; ++j) {
      const int n = n0 + wn * 64 + j * 16 + l16;
      b[j].x = V[(size_t)kk * DOUT + n];          // V is 256 KB -> L2 resident
      b[j].y = V[(size_t)(kk + 1) * DOUT + n];
    }
#pragma unroll
    for (int s = 0; s < 2; ++s)
#pragma unroll
      for (int j = 0; j < 4; ++j)
        acc[s][j] = wmma_f32(a[s], b[j], acc[s][j]);
  }

  // ---- bias + store ----
#pragma unroll
  for (int j = 0; j < 4; ++j) {
    const int n  = n0 + wn * 64 + j * 16 + l16;
    const float bv = bias[n];
#pragma unroll
    for (int s = 0; s < 2; ++s) {
      const int mbase = m0 + wm * 32 + s * 16 + kh * 8;
#pragma unroll
      for (int i = 0; i < 8; ++i)
        out[(size_t)(mbase + i) * DOUT + n] = acc[s][j][i] + bv;
    }
  }
}

extern "C" void kernel_launch(void* const* d_in, const int* in_sizes, int n_in,
                              void* d_out, int out_size, void* d_ws, size_t ws_size,
                              hipStream_t stream) {
  const float* x    = (const float*)d_in[0];
  const float* W    = (const float*)d_in[1];
  const float* U    = (const float*)d_in[2];
  const float* V    = (const float*)d_in[3];
  const float* bias = (const float*)d_in[4];
  float* out        = (float*)d_out;

  dim3 grid(DOUT / N_TILE, MROWS / M_TILE);   // 32 x 64 workgroups
  lora_linear_fused<<<grid, 256, 0, stream>>>(x, W, U, V, bias, out);
}